// GnrlPatSoftmaxRNN_46488726012391
// MI455X (gfx1250) — compile-verified
//
#include <hip/hip_runtime.h>

typedef __attribute__((ext_vector_type(2))) float v2f;
typedef __attribute__((ext_vector_type(4))) float v4f;
typedef __attribute__((ext_vector_type(8))) float v8f;

#define T_STEPS 256
#define BATCH 64
#define P_DIM 1024
#define H_DIM 512
#define DECAY_C 0.999f
#define MASK_FRAC_C 0.9f
#define TEMP_SCALE_C 10.0f

// One workgroup per batch. 1024 threads = 32 waves (wave32).
// Wave w exclusively owns pat rows [w*32, w*32+32) for both the WMMA GEMV
// and the rank-1 update, so ordinary same-wave global-memory ordering plus
// workgroup barriers (for LDS-staged softmax / h / new_h) are sufficient.
// pat state (128 MB) fits in the 192 MB L2 -> L2-resident across all 256
// timesteps; the kernel is L2-bandwidth bound, WMMA f32 keeps exact f32
// semantics for the GEMV.
__global__ __launch_bounds__(1024)
void GnrlPatSoftmaxRNN_kernel(const float* __restrict__ x,
                              float* __restrict__ patw,
                              float* __restrict__ out) {
  const int b      = blockIdx.x;
  const int tid    = threadIdx.x;
  const int wave   = tid >> 5;
  const int lane   = tid & 31;
  const int lane16 = lane & 15;   // M index within 16-row tile (A layout)
  const int khalf  = lane >> 4;   // 0: K=0..1, 1: K=2..3 (A layout)

  __shared__ __align__(16) float sh_h[H_DIM];
  __shared__ float sh_raw[P_DIM];
  __shared__ float sh_pr[P_DIM];
  __shared__ float sh_newh[H_DIM];
  __shared__ float sh_red[32];
  __shared__ float sh_scalar[2];

  float* patb = patw + (size_t)b * P_DIM * H_DIM;

  // B-operand mask: only lanes 0 and 16 carry the h column (N=0 of the 4x16
  // B matrix). Value-masking with cndmask/mul avoids EXEC save/restore in the
  // WMMA hot loop (EXEC must stay all-1s for WMMA anyway).
  const float bmask = (lane16 == 0) ? 1.0f : 0.0f;
  const bool  hi_k  = (khalf != 0);

  for (int t = 0; t < T_STEPS; ++t) {
    // ---- load h = x[t, b, :] into LDS ----
    if (tid < H_DIM)
      sh_h[tid] = x[(size_t)t * BATCH * H_DIM + (size_t)b * H_DIM + tid];
    __syncthreads();

    // ---- Phase 1: raw[p] = dot(pat[p,:], h) via V_WMMA_F32_16X16X4_F32 ----
    // A (16x4 f32): lane L<16 holds A[L, 0..1], lane 16+L holds A[L, 2..3].
    // B (4x16 f32): column 0 only -> lane 0 holds {h[k],h[k+1]},
    //               lane 16 holds {h[k+2],h[k+3]}, all other lanes zero.
    // C column 0: lane 0 VGPR r = M=r (0..7), lane 16 VGPR r = M=8+r.
#pragma unroll
    for (int tile = 0; tile < 2; ++tile) {
      const int p0 = wave * 32 + tile * 16;
      const float* arow = patb + (size_t)(p0 + lane16) * H_DIM + khalf * 2;
      v8f acc = {};
#pragma unroll 4
      for (int kb = 0; kb < H_DIM; kb += 4) {
        v2f a = *(const v2f*)(arow + kb);          // global_load_b64, coalesced-ish
        // Uniform-address LDS read -> bank broadcast, no divergence.
        v4f hq = *(const v4f*)(&sh_h[kb]);         // ds_load_b128
        v2f bv;
        bv.x = bmask * (hi_k ? hq.z : hq.x);       // v_cndmask + v_mul
        bv.y = bmask * (hi_k ? hq.w : hq.y);
        acc = __builtin_amdgcn_wmma_f32_16x16x4_f32(
            /*neg_a=*/false, a, /*neg_b=*/false, bv,
            /*c_mod=*/(short)0, acc, /*reuse_a=*/false, /*reuse_b=*/false);
      }
      if (lane16 == 0) {
        const int base = p0 + khalf * 8;
#pragma unroll
        for (int r = 0; r < 8; ++r) sh_raw[base + r] = acc[r];
      }
    }
    __syncthreads();

    // ---- Phase 2: masked softmax over P ----
    const float rv = sh_raw[tid];
    float mx = rv;
    for (int o = 16; o > 0; o >>= 1) mx = fmaxf(mx, __shfl_xor(mx, o, 32));
    if (lane == 0) sh_red[wave] = mx;
    if (tid >= 512) sh_newh[tid - 512] = 0.0f;  // zero new_h accumulator
    __syncthreads();
    if (wave == 0) {
      float v = sh_red[lane];
      for (int o = 16; o > 0; o >>= 1) v = fmaxf(v, __shfl_xor(v, o, 32));
      if (lane == 0) sh_scalar[0] = v;
    }
    __syncthreads();
    const float m    = sh_scalar[0];
    const float mask = (rv >= MASK_FRAC_C * m) ? 1.0f : 0.0f;
    const float e    = __expf(rv * mask / m * TEMP_SCALE_C);
    float s = e;
    for (int o = 16; o > 0; o >>= 1) s += __shfl_xor(s, o, 32);
    if (lane == 0) sh_red[wave] = s;
    __syncthreads();
    if (wave == 0) {
      float v = sh_red[lane];
      for (int o = 16; o > 0; o >>= 1) v += __shfl_xor(v, o, 32);
      if (lane == 0) sh_scalar[1] = v;
    }
    __syncthreads();
    sh_pr[tid] = e / sh_scalar[1];
    __syncthreads();

    // ---- Phase 3: new_h from OLD pat; rank-1 update + row L2 normalize ----
    float nhacc[16];
#pragma unroll
    for (int i = 0; i < 16; ++i) nhacc[i] = 0.0f;
    for (int r = 0; r < 32; ++r) {
      const int p     = wave * 32 + r;
      const float prv = sh_pr[p];
      float* row = patb + (size_t)p * H_DIM;
      float u[16];
      float sq = 0.0f;
#pragma unroll
      for (int i = 0; i < 16; ++i) {
        const int hh    = lane + 32 * i;           // coalesced, LDS conflict-free
        const float old = row[hh];
        const float un  = DECAY_C * old + prv * sh_h[hh];  // update_rate = 1
        u[i] = un;
        sq  += un * un;
        nhacc[i] += old * prv;                     // new_h uses pre-update pat
      }
      for (int o = 16; o > 0; o >>= 1) sq += __shfl_xor(sq, o, 32);
      const float inv = 1.0f / (sqrtf(sq) + 1e-10f);
#pragma unroll
      for (int i = 0; i < 16; ++i) row[lane + 32 * i] = u[i] * inv;
    }
#pragma unroll
    for (int i = 0; i < 16; ++i) atomicAdd(&sh_newh[lane + 32 * i], nhacc[i]);
    __syncthreads();
    if (tid < H_DIM)
      out[(size_t)t * BATCH * H_DIM + (size_t)b * H_DIM + tid] = sh_newh[tid];
    __syncthreads();
  }
}

extern "C" void kernel_launch(void* const* d_in, const int* in_sizes, int n_in,
                              void* d_out, int out_size, void* d_ws, size_t ws_size,
                              hipStream_t stream) {
  const float* x    = (const float*)d_in[0];  // [T, B, H]
  const float* pat0 = (const float*)d_in[1];  // [B, P, H]
  float* patw = (float*)d_ws;                 // mutable pat state (128 MB)

  const size_t pat_bytes = (size_t)BATCH * P_DIM * H_DIM * sizeof(float);
  // Inputs must not be mutated: copy initial pat into workspace each call
  // (async d2d copy is hipGraph-capture safe).
  hipMemcpyAsync(patw, pat0, pat_bytes, hipMemcpyDeviceToDevice, stream);

  GnrlPatSoftmaxRNN_kernel<<<BATCH, 1024, 0, stream>>>(x, patw, (float*)d_out);
}